// DeformableConv2d_22204980920785
// MI455X (gfx1250) — compile-verified
//
#include <hip/hip_runtime.h>
#include <math.h>

// Problem constants
#define BB    4
#define CC    64
#define OUTC  64
#define HH    128
#define WW    128
#define KKK   9           // K*K
#define KTOT  (CC * KKK)  // 576 GEMM depth
#define HW    (HH * WW)   // 16384
#define PIX   (BB * HW)   // 65536 total output pixels
#define NSTEP (KTOT / 4)  // 144 WMMA K-steps

// workspace layout (floats); all offsets 16B-aligned
#define WS_MASK  0                           // (B,9,H,W)              589824
#define WS_WQ1   (BB * KKK * HW)             // (144,32,4) lane-packed  18432
#define WS_WQ2   (WS_WQ1 + NSTEP * 32 * 4)   // (144,32,8) lane-packed  36864

typedef float v2f __attribute__((ext_vector_type(2)));
typedef float v8f __attribute__((ext_vector_type(8)));

// D = A(16x4 f32) * B(4x16 f32) + C(16x16 f32), CDNA5 wave32 WMMA
static __device__ inline v8f wmma4(v2f a, v2f b, v8f c) {
    return __builtin_amdgcn_wmma_f32_16x16x4_f32(
        /*neg_a=*/false, a, /*neg_b=*/false, b,
        /*c_mod=*/(short)0, c, /*reuse_a=*/false, /*reuse_b=*/false);
}

// ---------------------------------------------------------------------------
// Kernel 0: pack weights into per-lane WMMA B-fragment order.
// K is permuted kk-major: k' = kk*64 + c ; step s covers k' in [4s, 4s+4).
// For lane l (hl=l>>4, ln=l&15), element j of the B fragment is k'=4s+2*hl+j.
//   Wq1[s*128 + l*4 + {j, 2+j}]       = Wcat[n=ln | 16+ln][k']   (k_offmask)
//   Wq2[s*256 + l*8 + t*2 + j]        = w_dcn[n=t*16+ln][k']     (k_dcn)
// ---------------------------------------------------------------------------
__global__ __launch_bounds__(256) void k_prep(
    const float* __restrict__ w_off, const float* __restrict__ w_mask,
    const float* __restrict__ w_dcn,
    float* __restrict__ Wq1, float* __restrict__ Wq2)
{
    const int t = blockIdx.x * 256 + threadIdx.x;
    if (t < NSTEP * 32 * 8) {           // Wq2 element
        const int s  = t >> 8;
        const int r  = t & 255;
        const int l  = r >> 3;
        const int q  = r & 7;
        const int tt = q >> 1, j = q & 1;
        const int hl = l >> 4, ln = l & 15;
        const int kp = 4 * s + 2 * hl + j;
        const int kk = kp >> 6, c = kp & 63;
        Wq2[t] = w_dcn[(tt * 16 + ln) * KTOT + c * KKK + kk];
    }
    if (t < NSTEP * 32 * 4) {           // Wq1 element
        const int s  = t >> 7;
        const int r  = t & 127;
        const int l  = r >> 2;
        const int q  = r & 3;
        const int half = q >> 1, j = q & 1;
        const int hl = l >> 4, ln = l & 15;
        const int kp = 4 * s + 2 * hl + j;
        const int kk = kp >> 6, c = kp & 63;
        const int n  = half * 16 + ln;
        float v = 0.0f;
        if (n < 18)      v = w_off[n * KTOT + c * KKK + kk];
        else if (n < 27) v = w_mask[(n - 18) * KTOT + c * KKK + kk];
        Wq1[t] = v;
    }
}

// ---------------------------------------------------------------------------
// Kernel 1: offset conv (18ch) + mask conv (9ch, sigmoid), fused as one GEMM
//   M = 16 contiguous pixels (one wave per tile), N = 32 (27 used), K = 576
//   kk-outer: im2col address & pad mask hoisted; B = one b128 load per step.
// ---------------------------------------------------------------------------
__global__ __launch_bounds__(128) void k_offmask(
    const float* __restrict__ x,
    const float* __restrict__ Wq1,
    const float* __restrict__ b_off, const float* __restrict__ b_mask,
    float* __restrict__ out_off,      // (B,18,H,W) region inside d_out
    float* __restrict__ mask_ws)      // (B,9,H,W) scratch
{
    const int lane = threadIdx.x & 31;
    const int wave = threadIdx.x >> 5;
    const int tile = blockIdx.x * 4 + wave;
    const int m0   = tile << 4;
    const int b    = m0 >> 14;
    const int rem  = m0 & (HW - 1);
    const int y    = rem >> 7;
    const int x0   = rem & (WW - 1);
    const int hl   = lane >> 4;
    const int ln   = lane & 15;

    const float* xb = x + (size_t)b * CC * HW;
    const int px = x0 + ln;
    const float4* wq = (const float4*)Wq1 + lane;   // + s*32 per step

    v8f acc0 = {};   // N tile 0: channels 0..15  (offset)
    v8f acc1 = {};   // N tile 1: channels 16..31 (off 16-17, mask 18-26, pad)

#pragma unroll
    for (int kk = 0; kk < KKK; ++kk) {
        const int dh = kk / 3 - 1;
        const int dw = kk - (kk / 3) * 3 - 1;
        const int yy = y + dh;
        const int xx = px + dw;
        const int ok = (yy >= 0) & (yy < HH) & (xx >= 0) & (xx < WW);
        const float keep = ok ? 1.0f : 0.0f;
        const int yyc = min(max(yy, 0), HH - 1);
        const int xxc = min(max(xx, 0), WW - 1);
        const float* ap = xb + yyc * WW + xxc + 2 * hl * HW; // element j=0 addr

        for (int c0 = 0; c0 < CC; c0 += 4) {
            const int s = kk * 16 + (c0 >> 2);
            v2f a;
            a[0] = keep * ap[c0 * HW];
            a[1] = keep * ap[c0 * HW + HW];
            const float4 q = wq[s * 32];
            v2f bv0, bv1;
            bv0[0] = q.x; bv0[1] = q.y;
            bv1[0] = q.z; bv1[1] = q.w;
            acc0 = wmma4(a, bv0, acc0);
            acc1 = wmma4(a, bv1, acc1);
        }
    }

    // ---- epilogue: bias, sigmoid(mask), scatter ----
#pragma unroll
    for (int v = 0; v < 8; ++v) {
        const int m   = v + 8 * hl;
        const int pxx = x0 + m;
        out_off[((b * 18 + ln) * HH + y) * WW + pxx] = acc0[v] + b_off[ln];
        const int n1 = 16 + ln;
        if (n1 < 18) {
            out_off[((b * 18 + n1) * HH + y) * WW + pxx] = acc1[v] + b_off[n1];
        } else if (n1 < 27) {
            const int mm = n1 - 18;
            const float s = acc1[v] + b_mask[mm];
            mask_ws[((b * 9 + mm) * HH + y) * WW + pxx] =
                1.0f / (1.0f + expf(-s));
        }
    }
}

// ---------------------------------------------------------------------------
// Kernel 2: deformable conv main GEMM (kk-major K order)
//   per wave: 16 pixels x 64 outch (4 accumulators), K = 576.
//   Bilinear taps precomputed in LDS, loaded once per kk-block (16 steps).
//   B = two b128 loads per 4-WMMA step (lane-packed fragments).
// ---------------------------------------------------------------------------
__global__ __launch_bounds__(128) void k_dcn(
    const float* __restrict__ x,
    const float* __restrict__ Wq2, const float* __restrict__ b_dcn,
    const float* __restrict__ off,       // (B,18,H,W) from kernel 1
    const float* __restrict__ mask_ws,   // (B,9,H,W)
    float* __restrict__ out)             // (B,64,H,W)
{
    __shared__ float4 sW4[4][KKK][16];
    __shared__ int4   sI4[4][KKK][16];

    const int lane = threadIdx.x & 31;
    const int wave = threadIdx.x >> 5;
    const int tile = blockIdx.x * 4 + wave;
    const int m0   = tile << 4;
    const int b    = m0 >> 14;
    const int rem  = m0 & (HW - 1);
    const int y    = rem >> 7;
    const int x0   = rem & (WW - 1);
    const int hl   = lane >> 4;
    const int ln   = lane & 15;

    // ---- stage 1: per-(pixel,kk) bilinear taps into LDS ----
    for (int e = lane; e < 16 * KKK; e += 32) {
        const int ml  = e / KKK;
        const int kk  = e - ml * KKK;
        const int pxx = x0 + ml;
        const float dy = off[((b * 18 + 2 * kk)     * HH + y) * WW + pxx];
        const float dx = off[((b * 18 + 2 * kk + 1) * HH + y) * WW + pxx];
        const float mk = mask_ws[((b * 9 + kk) * HH + y) * WW + pxx];
        const int ki = kk / 3, kj = kk - ki * 3;
        const float py  = dy + (float)(ki + y - 1);
        const float pxf = dx + (float)(kj + pxx - 1);
        const float y0f = floorf(py), x0f = floorf(pxf);
        const float ly = py - y0f, lx = pxf - x0f;
        const int iy0 = (int)y0f, ix0 = (int)x0f;
        const int iy1 = iy0 + 1,  ix1 = ix0 + 1;
        const float vy0 = (iy0 >= 0 && iy0 < HH) ? 1.0f : 0.0f;
        const float vy1 = (iy1 >= 0 && iy1 < HH) ? 1.0f : 0.0f;
        const float vx0 = (ix0 >= 0 && ix0 < WW) ? 1.0f : 0.0f;
        const float vx1 = (ix1 >= 0 && ix1 < WW) ? 1.0f : 0.0f;
        const int cy0 = min(max(iy0, 0), HH - 1), cy1 = min(max(iy1, 0), HH - 1);
        const int cx0 = min(max(ix0, 0), WW - 1), cx1 = min(max(ix1, 0), WW - 1);
        sW4[wave][kk][ml] = make_float4((1.0f - ly) * (1.0f - lx) * mk * vy0 * vx0,
                                        (1.0f - ly) * lx          * mk * vy0 * vx1,
                                        ly          * (1.0f - lx) * mk * vy1 * vx0,
                                        ly          * lx          * mk * vy1 * vx1);
        sI4[wave][kk][ml] = make_int4(cy0 * WW + cx0, cy0 * WW + cx1,
                                      cy1 * WW + cx0, cy1 * WW + cx1);
    }
    __syncthreads();

    // ---- stage 2: K=576 GEMM, kk outer so taps load once per 16 steps ----
    const float* xb = x + (size_t)b * CC * HW;
    const float4* wq = (const float4*)Wq2 + lane * 2;  // + s*64 per step
    v8f acc[4] = {v8f{}, v8f{}, v8f{}, v8f{}};

    for (int kk = 0; kk < KKK; ++kk) {
        const float4 wv = sW4[wave][kk][ln];
        const int4   iv = sI4[wave][kk][ln];

        for (int c0 = 0; c0 < CC; c0 += 4) {
            const int s = kk * 16 + (c0 >> 2);
            v2f a;
#pragma unroll
            for (int j = 0; j < 2; ++j) {
                const float* pl = xb + (c0 + 2 * hl + j) * HW;
                a[j] = wv.x * pl[iv.x] + wv.y * pl[iv.y] +
                       wv.z * pl[iv.z] + wv.w * pl[iv.w];
            }
            const float4 qa = wq[s * 64];
            const float4 qb = wq[s * 64 + 1];
            v2f bv;
            bv[0] = qa.x; bv[1] = qa.y; acc[0] = wmma4(a, bv, acc[0]);
            bv[0] = qa.z; bv[1] = qa.w; acc[1] = wmma4(a, bv, acc[1]);
            bv[0] = qb.x; bv[1] = qb.y; acc[2] = wmma4(a, bv, acc[2]);
            bv[0] = qb.z; bv[1] = qb.w; acc[3] = wmma4(a, bv, acc[3]);
        }
    }

    // ---- epilogue: bias + store ----
#pragma unroll
    for (int t = 0; t < 4; ++t) {
        const int oc = t * 16 + ln;
#pragma unroll
        for (int v = 0; v < 8; ++v) {
            const int pxx = x0 + v + 8 * hl;
            out[((b * OUTC + oc) * HH + y) * WW + pxx] = acc[t][v] + b_dcn[oc];
        }
    }
}

// ---------------------------------------------------------------------------
extern "C" void kernel_launch(void* const* d_in, const int* in_sizes, int n_in,
                              void* d_out, int out_size, void* d_ws, size_t ws_size,
                              hipStream_t stream) {
    (void)in_sizes; (void)n_in; (void)out_size; (void)ws_size;
    const float* x      = (const float*)d_in[0];
    const float* w_off  = (const float*)d_in[1];
    const float* b_off  = (const float*)d_in[2];
    const float* w_mask = (const float*)d_in[3];
    const float* b_mask = (const float*)d_in[4];
    const float* w_dcn  = (const float*)d_in[5];
    const float* b_dcn  = (const float*)d_in[6];

    float* out     = (float*)d_out;                         // (B,64,H,W)
    float* off_out = out + (size_t)BB * OUTC * HW;          // (B,18,H,W)
    float* wsf     = (float*)d_ws;
    float* mask_ws = wsf + WS_MASK;                         // (B,9,H,W)
    float* Wq1     = wsf + WS_WQ1;                          // (144,32,4)
    float* Wq2     = wsf + WS_WQ2;                          // (144,32,8)

    const dim3 blk(128);                    // 4 waves, 4 pixel-tiles per block
    const dim3 grid(PIX / 16 / 4);          // 1024 blocks

    k_prep<<<dim3((NSTEP * 32 * 8 + 255) / 256), dim3(256), 0, stream>>>(
        w_off, w_mask, w_dcn, Wq1, Wq2);
    k_offmask<<<grid, blk, 0, stream>>>(x, Wq1, b_off, b_mask,
                                        off_out, mask_ws);
    k_dcn<<<grid, blk, 0, stream>>>(x, Wq2, b_dcn, off_out, mask_ws, out);
}